// Segment_79328045957851
// MI455X (gfx1250) — compile-verified
//
#include <hip/hip_runtime.h>
#include <hip/hip_bf16.h>

typedef __attribute__((ext_vector_type(2))) float    v2f;
typedef __attribute__((ext_vector_type(8))) float    v8f;
typedef __attribute__((ext_vector_type(4))) unsigned v4u;
typedef __attribute__((ext_vector_type(8))) unsigned v8u;

#define BATCH 128
#define IN_F  128
#define OUT_F 128
#define SEG   32
#define KDIM  (IN_F * SEG)   // 4096
#define KC    128            // K-chunk staged in LDS per TDM load
#define NCHUNK (KDIM / KC)   // 32

// ---------------------------------------------------------------------------
// Kernel 1: per (i,s,o) slope/intercept tables
// ---------------------------------------------------------------------------
__global__ void seg_prep_rc(const float* __restrict__ x,
                            const float* __restrict__ y,
                            float* __restrict__ R,
                            float* __restrict__ C) {
    int idx = blockIdx.x * blockDim.x + threadIdx.x;   // over IN_F*SEG*OUT_F
    if (idx >= IN_F * SEG * OUT_F) return;
    int o   = idx % OUT_F;
    int s   = (idx / OUT_F) % SEG;
    int i   = idx / (OUT_F * SEG);

    int base = (i * (SEG + 1) + s) * OUT_F + o;
    float x_lo = x[base];
    float x_hi = x[base + OUT_F];
    float y_lo = y[base];
    float y_hi = y[base + OUT_F];

    float div = x_hi - x_lo;
    div = (div == 0.0f) ? 1.0e-4f : div;
    float ratio = (y_hi - y_lo) / div;

    R[idx] = ratio;
    C[idx] = y_lo - ratio * x_lo;   // term = ratio*x_in + C
}

// ---------------------------------------------------------------------------
// Kernel 2: per (b,i) segment select -> one-hot expanded A-operands
// ---------------------------------------------------------------------------
__global__ void seg_select_pq(const float* __restrict__ x_in,
                              const float* __restrict__ x,
                              float* __restrict__ P,
                              float* __restrict__ Q) {
    int idx = blockIdx.x * blockDim.x + threadIdx.x;   // over BATCH*IN_F
    if (idx >= BATCH * IN_F) return;
    int i = idx % IN_F;
    int b = idx / IN_F;

    float xv = x_in[b * IN_F + i];
    float k_first = x[(i * (SEG + 1) + 0) * OUT_F];
    float k_last  = x[(i * (SEG + 1) + SEG) * OUT_F];
    bool below = xv <  k_first;
    bool above = xv >= k_last;

    float* Pb = P + b * KDIM + i * SEG;
    float* Qb = Q + b * KDIM + i * SEG;
#pragma unroll
    for (int s = 0; s < SEG; ++s) {
        float lo = x[(i * (SEG + 1) + s) * OUT_F];
        float hi = x[(i * (SEG + 1) + s + 1) * OUT_F];
        bool m = (xv >= lo) && (xv < hi);
        if (s == 0)       m = m || below;
        if (s == SEG - 1) m = m || above;
        Pb[s] = m ? xv : 0.0f;
        Qb[s] = m ? 1.0f : 0.0f;
    }
}

// ---------------------------------------------------------------------------
// TDM: async 2D tile load (KC rows x 16 f32 cols, row stride 128 f32) -> LDS.
// D# packing per CDNA5 ISA 8.3/8.4 (group0 128b, group1 256b); 2D tensor so
// groups 2/3 are omitted (VADDR2/3 = NULL).
// ---------------------------------------------------------------------------
__device__ __forceinline__ void tdm_load_tile(const float* gsrc, unsigned lds_addr) {
    unsigned long long ga = (unsigned long long)(const void*)gsrc;
    v4u g0;
    g0[0] = 1u;                                        // count=1, user mode
    g0[1] = lds_addr;                                  // lds_addr [63:32]
    g0[2] = (unsigned)(ga & 0xFFFFFFFFull);            // global_addr lo
    g0[3] = (unsigned)((ga >> 32) & 0x01FFFFFFull)     // global_addr [56:32]
          | 0x80000000u;                               // type=2 ("image")
    v8u g1;
    g1[0] = 0x00020000u;                               // data_size=2 (4B), no mask
    g1[1] = (16u << 16);                               // tensor_dim0 = 16
    g1[2] = ((unsigned)KC << 16);                      // tensor_dim1 = KC
    g1[3] = (16u << 16);                               // tile_dim0 = 16
    g1[4] = (unsigned)KC;                              // tile_dim1 = KC
    g1[5] = 128u;                                      // tensor_dim0_stride = 128
    g1[6] = 0u;
    g1[7] = 0u;
    asm volatile("tensor_load_to_lds %0, %1" :: "s"(g0), "s"(g1) : "memory");
}

// ---------------------------------------------------------------------------
// Kernel 3: out[b,o] = sum_k P[b,k]*R[k,o] + Q[b,k]*C[k,o]
// 8 blocks (one 16-col strip each) x 256 threads (8 wave32 = 8 row tiles).
// B panels (R,C) staged in LDS by TDM, double buffered; A panels from global.
// ---------------------------------------------------------------------------
__global__ void __launch_bounds__(256)
seg_wmma_gemm(const float* __restrict__ P,
              const float* __restrict__ Q,
              const float* __restrict__ R,
              const float* __restrict__ C,
              float* __restrict__ out) {
    __shared__ float smem[2 * 2 * KC * 16];            // [buf][R/C][KC][16] = 32 KB

    const int col0 = blockIdx.x * 16;
    const int wave = threadIdx.x >> 5;                 // 0..7 -> row tile
    const int row0 = wave * 16;
    const int lane = threadIdx.x & 31;
    const int half = lane >> 4;
    const int lr   = lane & 15;
    const int koff = half * 2;

    const bool wave0 = (wave == 0);
    const unsigned lds_base = (unsigned)(unsigned long long)(void*)smem;
    const unsigned buf_bytes = (unsigned)(KC * 16 * sizeof(float));

    const float* Prow = P + (row0 + lr) * KDIM;
    const float* Qrow = Q + (row0 + lr) * KDIM;

    // Prologue: issue chunk 0 into buffer 0 (R then C).
    if (wave0) {
        tdm_load_tile(R + col0, lds_base);
        tdm_load_tile(C + col0, lds_base + buf_bytes);
    }

    v8f acc = {};
    for (int c = 0; c < NCHUNK; ++c) {
        if (wave0) __builtin_amdgcn_s_wait_tensorcnt(0);  // chunk c landed
        __syncthreads();                                  // visible to all waves

        if (wave0 && (c + 1 < NCHUNK)) {                  // prefetch chunk c+1
            const int kn = (c + 1) * KC;
            const unsigned boff = ((unsigned)((c + 1) & 1)) * 2u * buf_bytes;
            tdm_load_tile(R + (size_t)kn * OUT_F + col0, lds_base + boff);
            tdm_load_tile(C + (size_t)kn * OUT_F + col0, lds_base + boff + buf_bytes);
        }

        const float* Rs = smem + (size_t)(c & 1) * 2 * KC * 16;
        const float* Cs = Rs + KC * 16;
        const int kbase = c * KC;

#pragma unroll 4
        for (int kk = 0; kk < KC; kk += 4) {
            v2f ap = *(const v2f*)(Prow + kbase + kk + koff);
            v2f aq = *(const v2f*)(Qrow + kbase + kk + koff);
            v2f br, bc;
            br.x = Rs[(kk + koff + 0) * 16 + lr];
            br.y = Rs[(kk + koff + 1) * 16 + lr];
            bc.x = Cs[(kk + koff + 0) * 16 + lr];
            bc.y = Cs[(kk + koff + 1) * 16 + lr];

            acc = __builtin_amdgcn_wmma_f32_16x16x4_f32(
                      false, ap, false, br, (short)0, acc, false, false);
            acc = __builtin_amdgcn_wmma_f32_16x16x4_f32(
                      false, aq, false, bc, (short)0, acc, false, false);
        }
        __syncthreads();   // done reading buf (c&1) before it is refilled at c+2
    }

#pragma unroll
    for (int r = 0; r < 8; ++r) {
        int m = half * 8 + r;
        out[(row0 + m) * OUT_F + col0 + lr] = acc[r];
    }
}

// ---------------------------------------------------------------------------
extern "C" void kernel_launch(void* const* d_in, const int* in_sizes, int n_in,
                              void* d_out, int out_size, void* d_ws, size_t ws_size,
                              hipStream_t stream) {
    const float* x_in = (const float*)d_in[0];  // (128,128)
    const float* x    = (const float*)d_in[1];  // (128,33,128)
    const float* y    = (const float*)d_in[2];  // (128,33,128)
    float* out = (float*)d_out;                 // (128,128)

    // Workspace: R, C (KDIM x OUT_F) and P, Q (BATCH x KDIM) — 8 MB total.
    float* R = (float*)d_ws;
    float* C = R + (size_t)KDIM * OUT_F;
    float* P = C + (size_t)KDIM * OUT_F;
    float* Q = P + (size_t)BATCH * KDIM;

    {
        int n = IN_F * SEG * OUT_F;                       // 524288
        seg_prep_rc<<<(n + 255) / 256, 256, 0, stream>>>(x, y, R, C);
    }
    {
        int n = BATCH * IN_F;                             // 16384
        seg_select_pq<<<(n + 255) / 256, 256, 0, stream>>>(x_in, x, P, Q);
    }
    {
        seg_wmma_gemm<<<OUT_F / 16, 256, 0, stream>>>(P, Q, R, C, out);
    }
}